// GNNEncoder_32134945309201
// MI455X (gfx1250) — compile-verified
//
#include <hip/hip_runtime.h>
#include <hip/hip_bf16.h>
#include <math.h>

typedef __attribute__((ext_vector_type(2))) float v2f;
typedef __attribute__((ext_vector_type(8))) float v8f;

#define CCH 128           // channels
#define BN_EPS 1e-5f

// ---------------------------------------------------------------- utilities

__global__ void zero_f4_kernel(float4* __restrict__ p, int n4) {
    int i = blockIdx.x * blockDim.x + threadIdx.x;
    if (i < n4) p[i] = make_float4(0.f, 0.f, 0.f, 0.f);
}

__global__ void count_kernel(const long long* __restrict__ dst,
                             float* __restrict__ cnt, int E) {
    int e = blockIdx.x * blockDim.x + threadIdx.x;
    if (e < E) atomicAdd(&cnt[(int)dst[e]], 1.0f);
}

__global__ void recip_kernel(float* __restrict__ inv, int n) {
    int i = blockIdx.x * blockDim.x + threadIdx.x;
    if (i < n) inv[i] = 1.0f / fmaxf(inv[i], 1.0f);
}

// transpose both weight matrices for the current layer: wT[n][k] = W[k][n]
__global__ void transpose2_kernel(const float* __restrict__ Wl,
                                  const float* __restrict__ Wr,
                                  float* __restrict__ WlT,
                                  float* __restrict__ WrT) {
    int idx = blockIdx.x * blockDim.x + threadIdx.x;   // 0..16383
    int k = idx >> 7;
    int n = idx & 127;
    WlT[n * CCH + k] = Wl[idx];
    WrT[n * CCH + k] = Wr[idx];
}

// ------------------------------------------------- edge aggregation (scatter)
// one wave32 per edge: lane l moves channels [4l, 4l+4)
__global__ __launch_bounds__(256)
void agg_kernel(const float* __restrict__ h,
                const long long* __restrict__ src,
                const long long* __restrict__ dst,
                float* __restrict__ agg, int E) {
    int wave = (int)((blockIdx.x * blockDim.x + threadIdx.x) >> 5);
    int lane = threadIdx.x & 31;
    if (wave >= E) return;
    long long s = src[wave];
    long long d = dst[wave];
    const float4 v = *(const float4*)(h + (size_t)s * CCH + lane * 4);
    float* a = agg + (size_t)d * CCH + lane * 4;
    atomicAdd(a + 0, v.x);
    atomicAdd(a + 1, v.y);
    atomicAdd(a + 2, v.z);
    atomicAdd(a + 3, v.w);
}

// ---------------------------------------- fused  mean@Wl + x@Wr + bias + BN (+mish)
// block = 256 threads = 8 waves; block owns 16 nodes x 128 out channels;
// wave w owns out-columns [16w, 16w+16). K=128 swept in steps of 4 with
// V_WMMA_F32_16X16X4_F32, both GEMMs folded into one f32 accumulator tile.
__global__ __launch_bounds__(256)
void sage_fused_gemm(const float* __restrict__ agg, const float* __restrict__ inv,
                     const float* __restrict__ hin,
                     const float* __restrict__ WlT, const float* __restrict__ WrT,
                     const float* __restrict__ bl,  const float* __restrict__ gamma,
                     const float* __restrict__ beta, const float* __restrict__ rmean,
                     const float* __restrict__ rvar, float* __restrict__ out,
                     int n_nodes, int apply_mish) {
    __shared__ float sM[16][132];   // mean tile (stride 132 avoids bank conflicts)
    __shared__ float sX[16][132];   // x tile

    const int t    = threadIdx.x;
    const int base = blockIdx.x * 16;

    // stage 16x128 mean (agg*inv) and x tiles into LDS; clamp (no divergence)
#pragma unroll
    for (int v = 0; v < 2; ++v) {
        int f    = t + v * 256;          // float4 chunk id, 0..511
        int row  = f >> 5;               // 0..15
        int c4   = (f & 31) * 4;         // 0..124
        int node = base + row;
        if (node >= n_nodes) node = n_nodes - 1;
        float4 m4 = *(const float4*)(agg + (size_t)node * CCH + c4);
        float4 x4 = *(const float4*)(hin + (size_t)node * CCH + c4);
        float s = inv[node];
        sM[row][c4 + 0] = m4.x * s;
        sM[row][c4 + 1] = m4.y * s;
        sM[row][c4 + 2] = m4.z * s;
        sM[row][c4 + 3] = m4.w * s;
        sX[row][c4 + 0] = x4.x;
        sX[row][c4 + 1] = x4.y;
        sX[row][c4 + 2] = x4.z;
        sX[row][c4 + 3] = x4.w;
    }
    __syncthreads();

    const int wave = t >> 5;
    const int lane = t & 31;
    const int m    = lane & 15;          // A row / D column index
    const int half = lane >> 4;          // K sub-slot (0: K+0..1, 1: K+2..3)
    const int n    = wave * 16 + m;      // output channel owned by this lane

    const float* wl = WlT + (size_t)n * CCH;
    const float* wr = WrT + (size_t)n * CCH;

    v8f acc = {};
#pragma unroll
    for (int k0 = 0; k0 < CCH; k0 += 4) {
        const int kk = k0 + half * 2;
        v2f a1 = *(const v2f*)&sM[m][kk];
        v2f b1 = *(const v2f*)(wl + kk);
        acc = __builtin_amdgcn_wmma_f32_16x16x4_f32(false, a1, false, b1,
                                                    (short)0, acc, false, false);
        v2f a2 = *(const v2f*)&sX[m][kk];
        v2f b2 = *(const v2f*)(wr + kk);
        acc = __builtin_amdgcn_wmma_f32_16x16x4_f32(false, a2, false, b2,
                                                    (short)0, acc, false, false);
    }

    // epilogue: + bias, BN(eval), optional mish
    const float bias = bl[n];
    const float scal = gamma[n] / sqrtf(rvar[n] + BN_EPS);
    const float mu   = rmean[n];
    const float bt   = beta[n];

#pragma unroll
    for (int i = 0; i < 8; ++i) {
        int row  = half * 8 + i;         // D layout: VGPR i -> M=i (lanes 0-15), M=8+i (lanes 16-31)
        int node = base + row;
        float val = acc[i] + bias;
        val = (val - mu) * scal + bt;
        if (apply_mish) {
            float sp = (val > 20.f) ? val : log1pf(expf(val));
            val = val * tanhf(sp);
        }
        if (node < n_nodes)
            out[(size_t)node * CCH + n] = val;
    }
}

// ---------------------------------------------------------------- launcher

extern "C" void kernel_launch(void* const* d_in, const int* in_sizes, int n_in,
                              void* d_out, int out_size, void* d_ws, size_t ws_size,
                              hipStream_t stream) {
    const float*     x     = (const float*)d_in[0];
    const long long* edge  = (const long long*)d_in[1];   // [2, E] int64
    const float*     Wl    = (const float*)d_in[2];       // [3,128,128]
    const float*     blv   = (const float*)d_in[3];       // [3,128]
    const float*     Wr    = (const float*)d_in[4];       // [3,128,128]
    const float*     gam   = (const float*)d_in[5];
    const float*     bet   = (const float*)d_in[6];
    const float*     rmean = (const float*)d_in[7];
    const float*     rvar  = (const float*)d_in[8];

    const int N  = in_sizes[0] / CCH;       // 100000
    const int E  = in_sizes[1] / 2;         // 1600000
    const size_t NC = (size_t)N * CCH;

    float* bufA = (float*)d_ws;             // activations between layers
    float* aggb = bufA + NC;                // neighbor sums
    float* invv = aggb + NC;                // 1/max(deg,1)
    float* wlT  = invv + N;                 // transposed Wl (current layer)
    float* wrT  = wlT + CCH * CCH;          // transposed Wr (current layer)

    const long long* esrc = edge;
    const long long* edst = edge + E;

    // degree counts -> reciprocal (layer-invariant, computed once)
    zero_f4_kernel<<<(N / 4 + 255) / 256, 256, 0, stream>>>((float4*)invv, N / 4);
    count_kernel<<<(E + 255) / 256, 256, 0, stream>>>(edst, invv, E);
    recip_kernel<<<(N + 255) / 256, 256, 0, stream>>>(invv, N);

    const int nTiles     = (N + 15) / 16;
    const int aggBlocks  = (int)(((size_t)E * 32 + 255) / 256);
    const int zeroBlocks = (int)((NC / 4 + 255) / 256);

    for (int layer = 0; layer < 3; ++layer) {
        const float* hin  = (layer == 0) ? x : bufA;
        float*       hout = (layer == 2) ? (float*)d_out : bufA;  // in-place safe

        zero_f4_kernel<<<zeroBlocks, 256, 0, stream>>>((float4*)aggb, (int)(NC / 4));
        agg_kernel<<<aggBlocks, 256, 0, stream>>>(hin, esrc, edst, aggb, E);

        transpose2_kernel<<<(CCH * CCH) / 256, 256, 0, stream>>>(
            Wl + (size_t)layer * CCH * CCH, Wr + (size_t)layer * CCH * CCH, wlT, wrT);

        sage_fused_gemm<<<nTiles, 256, 0, stream>>>(
            aggb, invv, hin, wlT, wrT,
            blv + layer * CCH, gam + layer * CCH, bet + layer * CCH,
            rmean + layer * CCH, rvar + layer * CCH,
            hout, N, (layer < 2) ? 1 : 0);
    }
}